// QuantumHybridNet_65481071400877
// MI455X (gfx1250) — compile-verified
//
#include <hip/hip_runtime.h>
#include <math.h>

typedef float v2f __attribute__((ext_vector_type(2)));
typedef float v8f __attribute__((ext_vector_type(8)));

#define BQ     2048
#define NPATCH 196
#define NW     4
#define NCLS   10
#define FTHRESH 0.95f
#define NTILE  13          // ceil(196/16)
#define QDIM   (NPATCH*NW) // 784

// ---------------------------------------------------------------------------
// Quantum circuit: 4-qubit state (16 complex amps) entirely in registers.
// Wire w -> bit mask 8>>w (wire 0 = MSB). MASK is a template constant so all
// state indexing is static -> stays in VGPRs (no scratch).
// ---------------------------------------------------------------------------
template<int MASK>
__device__ __forceinline__ void apply1q(float* sr, float* si,
    float u00r, float u00i, float u01r, float u01i,
    float u10r, float u10i, float u11r, float u11i) {
#pragma unroll
  for (int i = 0; i < 16; ++i) {
    if ((i & MASK) == 0) {
      const int j = i | MASK;
      float ar = sr[i], ai = si[i], br = sr[j], bi = si[j];
      sr[i] = u00r*ar - u00i*ai + u01r*br - u01i*bi;
      si[i] = u00r*ai + u00i*ar + u01r*bi + u01i*br;
      sr[j] = u10r*ar - u10i*ai + u11r*br - u11i*bi;
      si[j] = u10r*ai + u10i*ar + u11r*bi + u11i*br;
    }
  }
}
template<int MASK>
__device__ __forceinline__ void g_ry(float* sr, float* si, float t) {
  float s, c; sincosf(0.5f*t, &s, &c);
  apply1q<MASK>(sr, si, c,0.f, -s,0.f, s,0.f, c,0.f);
}
template<int MASK>
__device__ __forceinline__ void g_rx(float* sr, float* si, float t) {
  float s, c; sincosf(0.5f*t, &s, &c);
  apply1q<MASK>(sr, si, c,0.f, 0.f,-s, 0.f,-s, c,0.f);
}
template<int MASK>
__device__ __forceinline__ void g_rz(float* sr, float* si, float t) {
  float s, c; sincosf(0.5f*t, &s, &c);
  apply1q<MASK>(sr, si, c,-s, 0.f,0.f, 0.f,0.f, c,s);
}

__global__ void __launch_bounds__(256)
qcircuit_kernel(const float* __restrict__ x,
                const float* __restrict__ rang,
                const float* __restrict__ trx, const float* __restrict__ try_,
                const float* __restrict__ trz, const float* __restrict__ tcrx,
                float* __restrict__ qout) {
  const int tid = blockIdx.x * blockDim.x + threadIdx.x;
  if (tid >= BQ * NPATCH) return;
  const int b = tid / NPATCH, p = tid % NPATCH;
  const int pr = p / 14, pc = p % 14;
  const float* xb = x + (size_t)b * 784;
  const float d0 = xb[(2*pr+0)*28 + 2*pc+0];
  const float d1 = xb[(2*pr+0)*28 + 2*pc+1];
  const float d2 = xb[(2*pr+1)*28 + 2*pc+0];
  const float d3 = xb[(2*pr+1)*28 + 2*pc+1];

  float sr[16], si[16];
#pragma unroll
  for (int i = 0; i < 16; ++i) { sr[i] = 0.f; si[i] = 0.f; }
  sr[0] = 1.f;

  // RY encoding, wires 0..3
  g_ry<8>(sr, si, d0); g_ry<4>(sr, si, d1);
  g_ry<2>(sr, si, d2); g_ry<1>(sr, si, d3);
  // RandomLayer stand-in: gates rx/ry/rz cycling, wire i%4
  g_rx<8>(sr, si, rang[0]); g_ry<4>(sr, si, rang[1]);
  g_rz<2>(sr, si, rang[2]); g_rx<1>(sr, si, rang[3]);
  g_ry<8>(sr, si, rang[4]); g_rz<4>(sr, si, rang[5]);
  g_rx<2>(sr, si, rang[6]); g_ry<1>(sr, si, rang[7]);
  // variational singles
  g_rx<8>(sr, si, trx[0]);
  g_ry<4>(sr, si, try_[0]);
  g_rz<1>(sr, si, trz[0]);
  // CRX: control wire0 (bit 8), target wire2 (bit 2)
  {
    float s, c; sincosf(0.5f*tcrx[0], &s, &c);
#pragma unroll
    for (int i = 8; i < 16; ++i) {
      if ((i & 2) == 0) {
        const int j = i | 2;
        float ar = sr[i], ai = si[i], br = sr[j], bi = si[j];
        sr[i] = c*ar + s*bi;  si[i] = c*ai - s*br;
        sr[j] = s*ai + c*br;  si[j] = -s*ar + c*bi;
      }
    }
  }
  // Pauli-Z expectations per wire
  float e0 = 0.f, e1 = 0.f, e2 = 0.f, e3 = 0.f;
#pragma unroll
  for (int i = 0; i < 16; ++i) {
    const float pbb = sr[i]*sr[i] + si[i]*si[i];
    e0 += (i & 8) ? -pbb : pbb;
    e1 += (i & 4) ? -pbb : pbb;
    e2 += (i & 2) ? -pbb : pbb;
    e3 += (i & 1) ? -pbb : pbb;
  }
  float* q = qout + (size_t)tid * 4;
  q[0] = e0; q[1] = e1; q[2] = e2; q[3] = e3;
}

// ---------------------------------------------------------------------------
// fid/Laplacian: one wave per (batch, 16-row strip).
// wid is forced into an SGPR via readfirstlane so b/m/rowsFull and the
// Q/L base pointers are provably wave-uniform: the rowsFull branch lowers to
// a scalar branch and loads/stores use SADDR-based addressing.
// A tile loop-invariant; 13 WMMA_F32_16X16X4_F32 per wave. No diagonal
// checks inside the loop: diagonal entries are overwritten with the degree
// at the end (same-wave stores are ordered); the degree subtracts the
// self-loop term computed from the A fragment (shfl_xor 16).
// ---------------------------------------------------------------------------
__global__ void __launch_bounds__(256)
fid_lap_kernel(const float* __restrict__ qout, float* __restrict__ lap) {
  const int lane = threadIdx.x & 31;
  const int wid  = __builtin_amdgcn_readfirstlane(
      blockIdx.x * (blockDim.x >> 5) + (threadIdx.x >> 5));  // wave-uniform SGPR
  const int b = wid / NTILE;
  const int m = wid % NTILE;
  const int half = lane >> 4;      // 0: K=0,1   1: K=2,3
  const int l15  = lane & 15;

  const float* Q = qout + (size_t)b * QDIM;
  float* L = lap + (size_t)b * (NPATCH * NPATCH);

  // A tile: row = m*16 + l15, K pair by half
  const int rA = m * 16 + l15;
  const float maskA = (rA < NPATCH) ? 1.f : 0.f;
  const int rAc = (rA < NPATCH) ? rA : 0;
  v2f a = *(const v2f*)(Q + rAc * 4 + 2 * half);
  a.x *= maskA; a.y *= maskA;

  // self-loop weight for row rA: full dot = both K-halves via xor-16
  float part = a.x * a.x + a.y * a.y;
  const float selfdot = part + __shfl_xor(part, 16, 32);
  const float wself = (selfdot * selfdot >= FTHRESH) ? 1.f : 0.f; // row m*16+l15

  const bool rowsFull = (m < 12);          // scalar branch (m is SGPR)
  const int rowBase = m * 16 + half * 8;   // row of d[0] for this lane

  float rs[8];
#pragma unroll
  for (int v = 0; v < 8; ++v) rs[v] = 0.f;

  float* base = L + (size_t)(m * 16) * NPATCH + (size_t)(half * 8) * NPATCH + l15;

  // ---- interior column tiles: all columns valid; no diag handling ----
#pragma unroll
  for (int nt = 0; nt < 12; ++nt) {
    v2f bb = *(const v2f*)(Q + (nt * 16 + l15) * 4 + 2 * half);
    v8f cz = {};
    v8f d = __builtin_amdgcn_wmma_f32_16x16x4_f32(
        false, a, false, bb, (short)0, cz, false, false);

    float lv[8];
#pragma unroll
    for (int v = 0; v < 8; ++v)
      lv[v] = (d[v] * d[v] >= FTHRESH) ? -1.f : 0.f;
#pragma unroll
    for (int v = 0; v < 8; ++v) rs[v] -= lv[v];

    if (rowsFull) {
#pragma unroll
      for (int v = 0; v < 8; ++v) base[nt * 16 + v * NPATCH] = lv[v];
    } else {
#pragma unroll
      for (int v = 0; v < 8; ++v)
        if (rowBase + v < NPATCH) base[nt * 16 + v * NPATCH] = lv[v];
    }
  }

  // ---- tail column tile nt==12: cols 192..207, valid iff l15 < 4 ----
  {
    const int col = 192 + l15;
    const float maskB = (col < NPATCH) ? 1.f : 0.f;
    const int cBc = (col < NPATCH) ? col : 0;
    v2f bb = *(const v2f*)(Q + cBc * 4 + 2 * half);
    bb.x *= maskB; bb.y *= maskB;
    v8f cz = {};
    v8f d = __builtin_amdgcn_wmma_f32_16x16x4_f32(
        false, a, false, bb, (short)0, cz, false, false);

    float lv[8];
#pragma unroll
    for (int v = 0; v < 8; ++v)
      lv[v] = (d[v] * d[v] >= FTHRESH) ? -1.f : 0.f;
#pragma unroll
    for (int v = 0; v < 8; ++v) rs[v] -= lv[v];

    if (rowsFull) {
      if (l15 < 4) {
#pragma unroll
        for (int v = 0; v < 8; ++v) base[192 + v * NPATCH] = lv[v];
      }
    } else {
#pragma unroll
      for (int v = 0; v < 8; ++v)
        if ((rowBase + v < NPATCH) && (l15 < 4)) base[192 + v * NPATCH] = lv[v];
    }
  }

  // reduce connection counts across the 16 lanes of each half (butterfly)
#pragma unroll
  for (int v = 0; v < 8; ++v) {
    float r = rs[v];
    r += __shfl_xor(r, 1, 32);
    r += __shfl_xor(r, 2, 32);
    r += __shfl_xor(r, 4, 32);
    r += __shfl_xor(r, 8, 32);
    rs[v] = r;
  }
  // fetch w_self for row rowBase+v (held at lane l15 == 8*half+v, same in
  // both halves after the earlier xor-16), then patch diagonal with degree.
  float wsv[8];
#pragma unroll
  for (int v = 0; v < 8; ++v) wsv[v] = __shfl(wself, 8 * half + v, 32);
  if (l15 == 0) {
#pragma unroll
    for (int v = 0; v < 8; ++v) {
      const int row = rowBase + v;
      if (row < NPATCH) L[row * NPATCH + row] = rs[v] - wsv[v];
    }
  }
}

// ---------------------------------------------------------------------------
// Head: logits(B,10) = qout(B,784) @ head_w^T + head_b.
// 4 independent WMMA accumulator chains (784 = 16*49) to hide WMMA latency.
// ---------------------------------------------------------------------------
__global__ void __launch_bounds__(256)
head_kernel(const float* __restrict__ qout, const float* __restrict__ hw,
            const float* __restrict__ hb, float* __restrict__ logits) {
  const int lane = threadIdx.x & 31;
  const int wid  = __builtin_amdgcn_readfirstlane(
      blockIdx.x * (blockDim.x >> 5) + (threadIdx.x >> 5)); // 0..127, SGPR
  const int half = lane >> 4;
  const int l15  = lane & 15;
  const int row  = wid * 16 + l15;          // < 2048 always
  const float colMask = (l15 < NCLS) ? 1.f : 0.f;
  const int colC = (l15 < NCLS) ? l15 : 0;

  const float* qr = qout + (size_t)row * QDIM + 2 * half;
  const float* wr = hw + (size_t)colC * QDIM + 2 * half;

  v8f acc0 = {}, acc1 = {}, acc2 = {}, acc3 = {};
  for (int k = 0; k < QDIM; k += 16) {
    v2f a0 = *(const v2f*)(qr + k);
    v2f a1 = *(const v2f*)(qr + k + 4);
    v2f a2 = *(const v2f*)(qr + k + 8);
    v2f a3 = *(const v2f*)(qr + k + 12);
    v2f b0 = *(const v2f*)(wr + k);
    v2f b1 = *(const v2f*)(wr + k + 4);
    v2f b2 = *(const v2f*)(wr + k + 8);
    v2f b3 = *(const v2f*)(wr + k + 12);
    b0.x *= colMask; b0.y *= colMask;
    b1.x *= colMask; b1.y *= colMask;
    b2.x *= colMask; b2.y *= colMask;
    b3.x *= colMask; b3.y *= colMask;
    acc0 = __builtin_amdgcn_wmma_f32_16x16x4_f32(false, a0, false, b0, (short)0, acc0, false, false);
    acc1 = __builtin_amdgcn_wmma_f32_16x16x4_f32(false, a1, false, b1, (short)0, acc1, false, false);
    acc2 = __builtin_amdgcn_wmma_f32_16x16x4_f32(false, a2, false, b2, (short)0, acc2, false, false);
    acc3 = __builtin_amdgcn_wmma_f32_16x16x4_f32(false, a3, false, b3, (short)0, acc3, false, false);
  }
  v8f acc = (acc0 + acc1) + (acc2 + acc3);

  float* base = logits + (size_t)(wid * 16 + half * 8) * NCLS + l15;
  const float bias = hb[colC];
  if (l15 < NCLS) {
#pragma unroll
    for (int v = 0; v < 8; ++v) base[v * NCLS] = acc[v] + bias;
  }
}

// ---------------------------------------------------------------------------
// CNN encoder (output `encoded`): fused conv+relu+pool x2, then FC+relu.
// ---------------------------------------------------------------------------
__global__ void __launch_bounds__(256)
conv1_pool_kernel(const float* __restrict__ x, const float* __restrict__ w,
                  const float* __restrict__ bias, float* __restrict__ h1) {
  const int tid = blockIdx.x * blockDim.x + threadIdx.x;
  if (tid >= BQ * 8 * 14 * 14) return;
  int t = tid;
  const int xo = t % 14; t /= 14;
  const int yo = t % 14; t /= 14;
  const int co = t % 8;  const int b = t / 8;
  const float* xb = x + (size_t)b * 784;
  const float* wc = w + co * 9;
  const float bv = bias[co];
  float mx = 0.f;  // relu outputs are >= 0
#pragma unroll
  for (int py = 0; py < 2; ++py)
#pragma unroll
  for (int px = 0; px < 2; ++px) {
    const int oy = 2*yo + py, ox = 2*xo + px;
    float acc = bv;
#pragma unroll
    for (int ky = -1; ky <= 1; ++ky)
#pragma unroll
    for (int kx = -1; kx <= 1; ++kx) {
      const int iy = oy + ky, ix = ox + kx;
      const float val = (iy >= 0 && iy < 28 && ix >= 0 && ix < 28)
                        ? xb[iy*28 + ix] : 0.f;
      acc += val * wc[(ky+1)*3 + (kx+1)];
    }
    mx = fmaxf(mx, fmaxf(acc, 0.f));
  }
  h1[tid] = mx;
}

__global__ void __launch_bounds__(256)
conv2_pool_kernel(const float* __restrict__ h1, const float* __restrict__ w,
                  const float* __restrict__ bias, float* __restrict__ h2) {
  const int tid = blockIdx.x * blockDim.x + threadIdx.x;
  if (tid >= BQ * 16 * 7 * 7) return;
  int t = tid;
  const int xo = t % 7; t /= 7;
  const int yo = t % 7; t /= 7;
  const int co = t % 16; const int b = t / 16;
  const float* hbp = h1 + (size_t)b * 8 * 196;
  const float* wc = w + co * 72;
  const float bv = bias[co];
  float mx = 0.f;
#pragma unroll
  for (int py = 0; py < 2; ++py)
#pragma unroll
  for (int px = 0; px < 2; ++px) {
    const int oy = 2*yo + py, ox = 2*xo + px;
    float acc = bv;
    for (int ci = 0; ci < 8; ++ci) {
#pragma unroll
      for (int ky = -1; ky <= 1; ++ky)
#pragma unroll
      for (int kx = -1; kx <= 1; ++kx) {
        const int iy = oy + ky, ix = ox + kx;
        const float val = (iy >= 0 && iy < 14 && ix >= 0 && ix < 14)
                          ? hbp[ci*196 + iy*14 + ix] : 0.f;
        acc += val * wc[ci*9 + (ky+1)*3 + (kx+1)];
      }
    }
    mx = fmaxf(mx, fmaxf(acc, 0.f));
  }
  h2[tid] = mx;
}

__global__ void __launch_bounds__(256)
fc_kernel(const float* __restrict__ h2, const float* __restrict__ fcw,
          const float* __restrict__ fcb, float* __restrict__ enc) {
  const int tid = blockIdx.x * blockDim.x + threadIdx.x;
  if (tid >= BQ * 64) return;
  const int o = tid % 64, b = tid / 64;
  const float* hbp = h2 + (size_t)b * 784;
  const float* wo = fcw + (size_t)o * 784;
  float acc = fcb[o];
  for (int i = 0; i < 784; ++i) acc += hbp[i] * wo[i];
  enc[tid] = fmaxf(acc, 0.f);
}

// ---------------------------------------------------------------------------
extern "C" void kernel_launch(void* const* d_in, const int* in_sizes, int n_in,
                              void* d_out, int out_size, void* d_ws, size_t ws_size,
                              hipStream_t stream) {
  const float* x    = (const float*)d_in[0];
  const float* c1w  = (const float*)d_in[1];
  const float* c1b  = (const float*)d_in[2];
  const float* c2w  = (const float*)d_in[3];
  const float* c2b  = (const float*)d_in[4];
  const float* fcw  = (const float*)d_in[5];
  const float* fcb  = (const float*)d_in[6];
  const float* rang = (const float*)d_in[7];
  const float* trx  = (const float*)d_in[8];
  const float* try_ = (const float*)d_in[9];
  const float* trz  = (const float*)d_in[10];
  const float* tcrx = (const float*)d_in[11];
  const float* hw   = (const float*)d_in[12];
  const float* hb   = (const float*)d_in[13];

  float* out    = (float*)d_out;
  float* logits = out;                                         // B*10
  float* lap    = out + (size_t)BQ * NCLS;                     // B*196*196
  float* enc    = lap + (size_t)BQ * NPATCH * NPATCH;          // B*64

  float* ws   = (float*)d_ws;
  float* qout = ws;                                            // B*784
  float* h1   = qout + (size_t)BQ * QDIM;                      // B*8*14*14
  float* h2   = h1 + (size_t)BQ * 8 * 196;                     // B*16*7*7

  // quantum patch filter
  qcircuit_kernel<<<(BQ * NPATCH + 255) / 256, 256, 0, stream>>>(
      x, rang, trx, try_, trz, tcrx, qout);

  // graph Laplacian via WMMA f32 16x16x4 (2048*13 waves, 8 waves/block)
  fid_lap_kernel<<<(BQ * NTILE) / 8, 256, 0, stream>>>(qout, lap);

  // classifier head via WMMA (128 waves)
  head_kernel<<<16, 256, 0, stream>>>(qout, hw, hb, logits);

  // CNN encoder branch
  conv1_pool_kernel<<<(BQ * 8 * 196 + 255) / 256, 256, 0, stream>>>(x, c1w, c1b, h1);
  conv2_pool_kernel<<<(BQ * 16 * 49 + 255) / 256, 256, 0, stream>>>(h1, c2w, c2b, h2);
  fc_kernel<<<(BQ * 64 + 255) / 256, 256, 0, stream>>>(h2, fcw, fcb, enc);
}